// ColBlockQuantizedLinear_15702400434904
// MI455X (gfx1250) — compile-verified
//
#include <hip/hip_runtime.h>

typedef __attribute__((ext_vector_type(2))) float v2f;
typedef __attribute__((ext_vector_type(8))) float v8f;
typedef __attribute__((ext_vector_type(4))) int   v4i;

typedef v4i __attribute__((address_space(1))) g_v4i;   // global-AS int4
typedef v4i __attribute__((address_space(3))) l_v4i;   // LDS-AS int4

#define O_DIM   11008
#define I_DIM   4096
#define HALF_I  (I_DIM / 2)        // 2048 int32 per weight row
#define CHUNK_K 512                // K columns per staged chunk
#define CHUNK_HC (CHUNK_K / 2)     // 256 int32 per row per chunk
#define NCHUNK  (I_DIM / CHUNK_K)  // 8
#define ROW_PAD 260                // padded LDS row stride (ints): banks 4n+c, 16B aligned
#define WAVES   8
#define KPW     (CHUNK_K / WAVES)  // 64 K per wave per chunk
#define STEPS   (KPW / 4)          // 16 wmma (K=4) steps per wave per chunk
#define STAGE_OPS 4                // async b128 issues per thread per chunk

#if defined(__has_builtin)
#  if __has_builtin(__builtin_amdgcn_global_load_async_to_lds_b128)
#    define HAVE_ASYNC_LDS 1
#  endif
#  if __has_builtin(__builtin_amdgcn_s_wait_asynccnt)
#    define HAVE_WAIT_ASYNC 1
#  endif
#endif

#if HAVE_ASYNC_LDS
#  if HAVE_WAIT_ASYNC
#    define WAIT_ASYNC(n) __builtin_amdgcn_s_wait_asynccnt(n)
#  else
#    define WAIT_ASYNC(n) asm volatile("s_wait_asynccnt %0" ::"i"(n))
#  endif
#else
#  define WAIT_ASYNC(n) ((void)0)
#endif

// Stage one 16-row x 256-int32 weight chunk into (padded) LDS.
// Coalesced: 64 consecutive threads cover one contiguous 1KB row segment.
__device__ __forceinline__ void stage_chunk(const int* __restrict__ gsrc,
                                            int* __restrict__ lbase, int tid) {
#pragma unroll
    for (int it = 0; it < STAGE_OPS; ++it) {
        int f   = tid + 256 * it;        // int4 index 0..1023
        int r   = f >> 6;                // row 0..15
        int col = (f & 63) << 2;         // int32 col, 16B aligned
#if HAVE_ASYNC_LDS
        __builtin_amdgcn_global_load_async_to_lds_b128(
            (g_v4i*)(void*)(gsrc + r * HALF_I + col),
            (l_v4i*)(lbase + r * ROW_PAD + col),
            0, 0);
#else
        int4 v = *(const int4*)(gsrc + r * HALF_I + col);
        *(int4*)&lbase[r * ROW_PAD + col] = v;
#endif
    }
}

__global__ __launch_bounds__(256)
void q4_linear_wmma_f32(const float* __restrict__ inp,
                        const int*   __restrict__ qw,
                        const float* __restrict__ scales,
                        const float* __restrict__ zeros,
                        float*       __restrict__ out) {
    __shared__ int smem[2][16 * ROW_PAD];   // 2 x 16.6 KB, double-buffered staging

    const int tid  = threadIdx.x;
    const int wave = tid >> 5;
    const int lane = tid & 31;
    const int n    = lane & 15;          // output column within tile / A row (M)
    const int g    = lane >> 4;          // half-wave group (K split)
    const int tile = blockIdx.x;         // 688 tiles of 16 outputs

    const int   orow = tile * 16 + n;
    const float sc   = scales[orow];
    const float nzs  = -zeros[orow] * sc;        // fold (w - z)*s = fma(w, s, -z*s)
    const float* arow = inp + (size_t)n * I_DIM; // A-matrix row M = n for this lane

    v8f acc = {};

    const int* gq = qw + (size_t)tile * 16 * HALF_I;

    // Prologue: kick off chunk 0 into buffer 0.
    stage_chunk(gq, smem[0], tid);

    for (int c = 0; c < NCHUNK; ++c) {
        // Issue DMA for the next chunk into the other buffer (its previous
        // contents, chunk c-1, were retired at the trailing barrier below).
        if (c + 1 < NCHUNK) {
            stage_chunk(gq + (c + 1) * CHUNK_HC, smem[(c + 1) & 1], tid);
            WAIT_ASYNC(STAGE_OPS);   // in-order: chunk c's 4 transfers are done
        } else {
            WAIT_ASYNC(0);
        }
        __syncthreads();             // all waves' chunk-c data visible in LDS

        // ---- this wave: K slice [c*512 + wave*64, +64) as 16 WMMA(16x16x4) ----
        const int* sbuf = smem[c & 1];
        const int  hcb  = wave * (KPW / 2) + g;             // half-col base in chunk
        const int  kb   = c * CHUNK_K + wave * KPW + 2 * g; // global K base for A
#pragma unroll
        for (int s = 0; s < STEPS; ++s) {
            int   q  = sbuf[n * ROW_PAD + hcb + 2 * s];  // one byte: 2 nibbles = K pair
            float b0 = fmaf((float)(q & 15),        sc, nzs); // K = 4s+2g   (low nibble)
            float b1 = fmaf((float)((q >> 4) & 15), sc, nzs); // K = 4s+2g+1 (high nibble)
            v2f b; b[0] = b0; b[1] = b1;
            v2f a = *(const v2f*)(arow + kb + 4 * s);    // A[m][4s+2g .. +1]
            acc = __builtin_amdgcn_wmma_f32_16x16x4_f32(
                      false, a, false, b, (short)0, acc, false, false);
        }
        __syncthreads();   // chunk-c buffer free before iter c+1 DMAs chunk c+2 into it
    }

    // ---- cross-wave reduction of 8 partial 16x16 tiles (LDS reuse) ----
    float* red = (float*)&smem[0][0];
#pragma unroll
    for (int j = 0; j < 8; ++j)
        red[wave * 256 + lane * 8 + j] = ((float*)&acc)[j];
    __syncthreads();

    float sum = 0.0f;
#pragma unroll
    for (int w = 0; w < WAVES; ++w)
        sum += red[w * 256 + tid];

    // element id tid = l*8 + j : C layout VGPR j, lane l -> M = j + 8*(l>>4), N = l&15
    int l  = tid >> 3;
    int j  = tid & 7;
    int mm = j + ((l >> 4) << 3);
    int nn = l & 15;
    out[(size_t)mm * O_DIM + tile * 16 + nn] = sum;
}

extern "C" void kernel_launch(void* const* d_in, const int* in_sizes, int n_in,
                              void* d_out, int out_size, void* d_ws, size_t ws_size,
                              hipStream_t stream) {
    const float* inp    = (const float*)d_in[0];
    const int*   qw     = (const int*)d_in[1];
    const float* scales = (const float*)d_in[2];
    const float* zeros  = (const float*)d_in[3];
    float*       out    = (float*)d_out;
    (void)in_sizes; (void)n_in; (void)out_size; (void)d_ws; (void)ws_size;

    dim3 grid(O_DIM / 16);   // 688 tiles
    dim3 block(256);         // 8 wave32s
    hipLaunchKernelGGL(q4_linear_wmma_f32, grid, block, 0, stream,
                       inp, qw, scales, zeros, out);
}